// SepConv_8924942041978
// MI455X (gfx1250) — compile-verified
//
#include <hip/hip_runtime.h>

// Problem constants (from reference setup_inputs)
#define BATCH 8
#define CH    3
#define WIMG  512
#define KSZ   13
#define WO    500          // 512 - 13 + 1

// Tiling
#define TI    32           // output rows per block
#define TJ    32           // output cols per block
#define RS    44           // staged rows  = TI + KSZ - 1
#define QS    11           // 11 x b128 (16B) chunks = 44 floats per staged row
#define LS    48           // LDS row stride in floats (16B-aligned, conflict-free)

__global__ __launch_bounds__(256)
void sepconv_kernel(const float* __restrict__ img,
                    const float* __restrict__ hori,
                    const float* __restrict__ vert,
                    float* __restrict__ out)
{
    __shared__ float smem[CH * RS * LS];   // 3 * 44 * 48 * 4 = 25,344 bytes

    const int tid = threadIdx.x;
    const int b   = blockIdx.z;
    const int i0  = blockIdx.y * TI;
    const int j0  = blockIdx.x * TJ;

    // ---------------------------------------------------------------
    // Stage img tile (3 channels x 44 x 44 halo) into LDS using the
    // CDNA5 async global->LDS DMA path (tracked by ASYNCcnt).
    // Each lane issues 16-byte chunks; 11 chunks cover a 44-float row.
    // Out-of-image rows/cols are clamped inside the buffer; the data
    // they deliver only feeds store-masked output pixels.
    //
    // NOTE: &smem[0] is passed as a (template-unused) operand so the
    // shared array's address ESCAPES into the asm. Combined with the
    // "memory" clobber this forces LLVM to treat the asm as a writer
    // of smem — otherwise every ds_load after the barrier folds to
    // undef and the whole compute is dead-code eliminated (observed
    // in round 1: ds=0).
    // ---------------------------------------------------------------
    const float* gbase = img + (size_t)b * CH * WIMG * WIMG;  // wave-uniform SGPR pair
    const int NCHUNK = CH * RS * QS;                          // 1452
    for (int t = tid; t < NCHUNK; t += 256) {
        int c   = t / (RS * QS);
        int rem = t - c * (RS * QS);
        int r   = rem / QS;
        int q   = rem - r * QS;
        int gr  = i0 + r;      if (gr > WIMG - 1) gr = WIMG - 1;
        int gc  = j0 + q * 4;  if (gc > WIMG - 4) gc = WIMG - 4;
        unsigned gOff = (unsigned)(((c * WIMG + gr) * WIMG + gc) * 4);
        unsigned lOff = (unsigned)(((c * RS + r) * LS + q * 4) * 4);
        // GVS mode: mem_addr = SADDR(sgpr64) + VADDR(u32); VDST vgpr = LDS byte addr
        asm volatile("global_load_async_to_lds_b128 %0, %1, %2"
                     :: "v"(lOff), "v"(gOff), "s"(gbase), "v"(&smem[0])
                     : "memory");
    }
    asm volatile("s_wait_asynccnt 0" ::: "memory");
    __syncthreads();

    // ---------------------------------------------------------------
    // Per-thread: 4 output rows x 1 column, all 3 channels.
    // Lanes 0..31 = consecutive j -> coalesced hori/vert/out traffic.
    // ---------------------------------------------------------------
    const int tx = tid & 31;            // column within tile
    const int ty = tid >> 5;            // row group (0..7)
    const int ro = ty * 4;              // first local output row
    const int jl = tx;                  // local col
    const int j  = j0 + jl;
    const int jc = (j < WO) ? j : (WO - 1);

    // Per-pixel weights: 4 rows x 13 taps x 2 -> kept in VGPRs.
    float hw[4][KSZ];
    float vw[4][KSZ];
#pragma unroll
    for (int rr = 0; rr < 4; ++rr) {
        int i  = i0 + ro + rr;
        int ic = (i < WO) ? i : (WO - 1);
#pragma unroll
        for (int v = 0; v < KSZ; ++v) {
            size_t idx = (((size_t)b * KSZ + v) * WO + ic) * (size_t)WO + jc;
            hw[rr][v] = hori[idx];
            vw[rr][v] = vert[idx];
        }
    }

    // out[i,j] = sum_u vw[u] * ( sum_v img[i+u][j+v] * hw[v] )
    // Register-block 4 rows: each staged img row (13 LDS reads) is reused by
    // up to 4 output rows -> 52 instead of 169 LDS b32 reads per pixel/channel.
#pragma unroll 1
    for (int c = 0; c < CH; ++c) {
        float acc[4] = {0.f, 0.f, 0.f, 0.f};
#pragma unroll
        for (int rIm = 0; rIm < 16; ++rIm) {          // 4 rows + 12 halo
            float row[KSZ];
            const int sbase = (c * RS + ro + rIm) * LS + jl;
#pragma unroll
            for (int v = 0; v < KSZ; ++v)
                row[v] = smem[sbase + v];
#pragma unroll
            for (int rr = 0; rr < 4; ++rr) {
                const int u = rIm - rr;               // compile-time after unroll
                if (u >= 0 && u < KSZ) {
                    float s = 0.f;
#pragma unroll
                    for (int v = 0; v < KSZ; ++v)
                        s = fmaf(row[v], hw[rr][v], s);
                    acc[rr] = fmaf(s, vw[rr][u], acc[rr]);
                }
            }
        }
#pragma unroll
        for (int rr = 0; rr < 4; ++rr) {
            int i = i0 + ro + rr;
            if (i < WO && j < WO)
                out[(((size_t)b * CH + c) * WO + i) * (size_t)WO + j] = acc[rr];
        }
    }
}

extern "C" void kernel_launch(void* const* d_in, const int* in_sizes, int n_in,
                              void* d_out, int out_size, void* d_ws, size_t ws_size,
                              hipStream_t stream) {
    (void)in_sizes; (void)n_in; (void)d_ws; (void)ws_size; (void)out_size;
    const float* img  = (const float*)d_in[0];
    const float* hori = (const float*)d_in[1];
    const float* vert = (const float*)d_in[2];
    float* out = (float*)d_out;

    dim3 grid((WO + TJ - 1) / TJ,   // 16
              (WO + TI - 1) / TI,   // 16
              BATCH);               // 8
    sepconv_kernel<<<grid, dim3(256), 0, stream>>>(img, hori, vert, out);
}